// BaseMetricS2_37125697307043
// MI455X (gfx1250) — compile-verified
//
#include <hip/hip_runtime.h>

// ---------------------------------------------------------------------------
// BaseMetricS2: weighted micro confusion-matrix means on the sphere.
// pred  [8,16,721,1440] f32, truth [8,721,1440] i32, qw [721,1440] f32
// out   [4] f32 = (mean tp, mean fp, mean fn, mean tn) over [B,C] bins.
//
// Memory-bound streaming reduction (568 MB @ 23.3 TB/s ~= 24 us floor).
// Only 3 global sums (tp/fp/fn) + S = sum(qw) are needed:
//   mean(x) over [B,C] = sum(x) / (B*C);  mean(tn) = S - the other three means.
// Grid is 2D: blockIdx.y = batch (kills the per-thread integer division),
// blockIdx.x tiles the pixel quads within one batch.
// ---------------------------------------------------------------------------

#define TPB 256
#define QPT 4 // quads (of 4 pixels) per thread

constexpr int kB = 8;
constexpr int kC = 16;
constexpr int kNLAT = 721;
constexpr int kNLON = 1440;
constexpr int kN = kNLAT * kNLON;          // 1,038,240 pixels (divisible by 4)
constexpr int kQuadsPerB = kN / 4;         // 259,560 float4-quads per batch
constexpr int kQPB = TPB * QPT;            // 1024 quads per block
constexpr int kGridX = (kQuadsPerB + kQPB - 1) / kQPB;  // 254
constexpr int kIgnore = -100;

__global__ __launch_bounds__(TPB) void metric_main_kernel(
    const float* __restrict__ pred, const int* __restrict__ truth,
    const float* __restrict__ qw, float* __restrict__ acc) {
  const int b = (int)blockIdx.y;                       // batch index (scalar)
  const float* __restrict__ predb = pred + (size_t)(b * kC) * (size_t)kN;
  const int* __restrict__ truthb = truth + (size_t)b * (size_t)kN;

  float tp = 0.f, fp = 0.f, fn = 0.f, wsum = 0.f;

#pragma unroll
  for (int i = 0; i < QPT; ++i) {
    int q = (int)blockIdx.x * kQPB + i * TPB + (int)threadIdx.x;
    if (q < kQuadsPerB) {
      int p = q * 4;                                   // pixel within batch

      const float* basep = predb + (size_t)p;

      // 16-class argmax, first-index-wins (strict '>'), 4 pixels at once.
      float4 best = *(const float4*)basep;
      int c0 = 0, c1 = 0, c2 = 0, c3 = 0;
#pragma unroll
      for (int c = 1; c < kC; ++c) {
        const float* cp = basep + (size_t)c * (size_t)kN;
        // Prefetch this class-plane stream two i-iterations ahead
        // -> global_prefetch_b8; speculative, OOB prefetches are dropped.
        __builtin_prefetch(cp + 2 * TPB * 4, 0, 1);
        float4 v = *(const float4*)cp;  // global_load_b128, coalesced per plane
        if (v.x > best.x) { best.x = v.x; c0 = c; }
        if (v.y > best.y) { best.y = v.y; c1 = c; }
        if (v.z > best.z) { best.z = v.z; c2 = c; }
        if (v.w > best.w) { best.w = v.w; c3 = c; }
      }

      int4 t = *(const int4*)(truthb + p);
      float4 w = *(const float4*)(qw + p);

      // Exact reference semantics:
      //  ignore(-100) -> both labels -1 -> matched, but invalid -> contributes 0
      //  tp += w if valid(t) &&  matched ; fn += w if valid(t) && !matched
      //  fp += w if valid(o) && !matched  (o always in [0,C) unless ignored)
      auto acc_elem = [&](int tt, int oo, float wv) {
        bool ig = (tt == kIgnore);
        int tl = ig ? -1 : tt;
        int ol = ig ? -1 : oo;
        bool m = (tl == ol);
        bool vt = ((unsigned)tl < (unsigned)kC);
        bool vo = ((unsigned)ol < (unsigned)kC);
        if (vt) { if (m) tp += wv; else fn += wv; }
        if (vo && !m) fp += wv;
      };
      acc_elem(t.x, c0, w.x);
      acc_elem(t.y, c1, w.y);
      acc_elem(t.z, c2, w.z);
      acc_elem(t.w, c3, w.w);

      if (b == 0) wsum += w.x + w.y + w.z + w.w;  // S = sum(qw), counted once
    }
  }

  // wave32 tree reduction
#pragma unroll
  for (int off = 16; off > 0; off >>= 1) {
    tp   += __shfl_down(tp,   off, 32);
    fp   += __shfl_down(fp,   off, 32);
    fn   += __shfl_down(fn,   off, 32);
    wsum += __shfl_down(wsum, off, 32);
  }

  __shared__ float sred[4];
  if (threadIdx.x < 4) sred[threadIdx.x] = 0.f;
  __syncthreads();
  if ((threadIdx.x & 31u) == 0u) {  // one leader per wave32
    atomicAdd(&sred[0], tp);
    atomicAdd(&sred[1], fp);
    atomicAdd(&sred[2], fn);
    atomicAdd(&sred[3], wsum);
  }
  __syncthreads();
  if (threadIdx.x == 0) {
    atomicAdd(&acc[0], sred[0]);
    atomicAdd(&acc[1], sred[1]);
    atomicAdd(&acc[2], sred[2]);
    atomicAdd(&acc[3], sred[3]);
  }
}

__global__ void metric_init_kernel(float* __restrict__ acc) {
  if (threadIdx.x < 4) acc[threadIdx.x] = 0.f;
}

__global__ void metric_finalize_kernel(const float* __restrict__ acc,
                                       float* __restrict__ out) {
  if (threadIdx.x == 0) {
    const float inv_bc = 1.0f / (float)(kB * kC);
    float tp = acc[0], fp = acc[1], fn = acc[2], S = acc[3];
    out[0] = tp * inv_bc;                       // mean tp
    out[1] = fp * inv_bc;                       // mean fp
    out[2] = fn * inv_bc;                       // mean fn
    out[3] = S - (tp + fp + fn) * inv_bc;       // mean tn = S - others
  }
}

extern "C" void kernel_launch(void* const* d_in, const int* in_sizes, int n_in,
                              void* d_out, int out_size, void* d_ws, size_t ws_size,
                              hipStream_t stream) {
  const float* pred  = (const float*)d_in[0];
  const int*   truth = (const int*)d_in[1];
  const float* qw    = (const float*)d_in[2];
  float* acc = (float*)d_ws;
  float* out = (float*)d_out;

  metric_init_kernel<<<1, 32, 0, stream>>>(acc);
  dim3 grid(kGridX, kB, 1);
  metric_main_kernel<<<grid, TPB, 0, stream>>>(pred, truth, qw, acc);
  metric_finalize_kernel<<<1, 32, 0, stream>>>(acc, out);
}